// SRNN_6519760355473
// MI455X (gfx1250) — compile-verified
//
#include <hip/hip_runtime.h>
#include <hip/hip_bf16.h>
#include <math.h>

#define ALPHA 0.8f
#define KAPPA 0.8f
#define THR   1.0f
#define NB    1000
#define NI    784
#define NR    256
#define NO    10
#define SPLITS 4          // b-split for streaming reductions (1000/4 = 250 each)

typedef __attribute__((ext_vector_type(2))) float v2f;
typedef __attribute__((ext_vector_type(8))) float v8f;

// ---------------------------------------------------------------------------
// Elementwise trace update: out = alpha*A + B
// ---------------------------------------------------------------------------
__global__ void axpb_kernel(const float* __restrict__ A, const float* __restrict__ B,
                            float alpha, float* __restrict__ out, int n) {
    int i = blockIdx.x * blockDim.x + threadIdx.x;
    if (i < n) out[i] = alpha * A[i] + B[i];
}

// ---------------------------------------------------------------------------
// Forward: v_new = ALPHA*v*(1-z) + z@w_rec_eff.T + x@w_in.T  (fp32 WMMA)
// Emits h = max(0, 1-|v_new-1|) and z_new = (v_new > 1).
// One wave per 16x16 (batch x rec) tile; K-loop in steps of 4.
// Out-of-range batch rows are CLAMPED (not masked): they only affect C rows
// that the epilogue never stores, so the K-loop has zero divergence and
// every operand is a single unconditional global_load_b64.
// ---------------------------------------------------------------------------
__global__ __launch_bounds__(128) void fwd_wmma_kernel(
    const float* __restrict__ x, const float* __restrict__ z,
    const float* __restrict__ v, const float* __restrict__ w_in,
    const float* __restrict__ w_rec,
    float* __restrict__ h_out, float* __restrict__ zf_out)
{
    const int NTN = NR / 16;                                    // 16 N-tiles
    int wave = blockIdx.x * (blockDim.x >> 5) + (threadIdx.x >> 5);
    int tm = wave / NTN, tn = wave % NTN;
    int lane = threadIdx.x & 31;
    int half = lane >> 4;
    int mr   = lane & 15;
    int mA = tm * 16 + mr;          // batch row feeding A operand
    if (mA >= NB) mA = NB - 1;      // clamp: garbage rows are discarded later
    int nB = tn * 16 + mr;          // rec unit feeding B operand

    const float* xrow = x     + (size_t)mA * NI + 2 * half;
    const float* wrow = w_in  + (size_t)nB * NI + 2 * half;
    const float* zrow = z     + (size_t)mA * NR + 2 * half;
    const float* rrow = w_rec + (size_t)nB * NR + 2 * half;

    v8f c = {};
    // --- x @ w_in.T ---
    #pragma unroll 4
    for (int k0 = 0; k0 < NI; k0 += 4) {
        v2f a = *(const v2f*)(xrow + k0);
        v2f b = *(const v2f*)(wrow + k0);
        c = __builtin_amdgcn_wmma_f32_16x16x4_f32(false, a, false, b,
                                                  (short)0, c, false, false);
    }
    // --- z @ w_rec_eff.T (diagonal of w_rec zeroed via cndmask, no branch) ---
    #pragma unroll 4
    for (int k0 = 0; k0 < NR; k0 += 4) {
        int k = k0 + 2 * half;
        v2f a = *(const v2f*)(zrow + k0);
        v2f b = *(const v2f*)(rrow + k0);
        b.x = (nB == k)     ? 0.0f : b.x;
        b.y = (nB == k + 1) ? 0.0f : b.y;
        c = __builtin_amdgcn_wmma_f32_16x16x4_f32(false, a, false, b,
                                                  (short)0, c, false, false);
    }
    // --- epilogue: leak term, pseudo-derivative, spike ---
    #pragma unroll
    for (int g = 0; g < 8; ++g) {
        int bi = tm * 16 + g + 8 * half;     // C/D row: M = g + 8*half
        int r  = tn * 16 + mr;               // C/D col: N = lane&15
        if (bi < NB) {
            size_t off = (size_t)bi * NR + r;
            float vold = v[off], zold = z[off];
            float vn = c[g] + ALPHA * vold * (1.0f - zold);
            float hh = 1.0f - fabsf(vn - THR);        // GAMMA = 1
            h_out[off]  = hh > 0.0f ? hh : 0.0f;
            zf_out[off] = (vn > THR) ? 1.0f : 0.0f;
        }
    }
}

// ---------------------------------------------------------------------------
// vo_new = KAPPA*vo + z_new@w_out.T ; yo = softmax(vo_new) ; err = yo - yt
// One thread per batch row (O=10 is tiny).
// ---------------------------------------------------------------------------
__global__ void softmax_err_kernel(const float* __restrict__ zf,
                                   const float* __restrict__ vo,
                                   const float* __restrict__ w_out,
                                   const float* __restrict__ yt,
                                   float* __restrict__ yo_out,
                                   float* __restrict__ err_out)
{
    int b = blockIdx.x * blockDim.x + threadIdx.x;
    if (b >= NB) return;
    const float* zr = zf + (size_t)b * NR;
    float von[NO];
    #pragma unroll
    for (int o = 0; o < NO; ++o) {
        float acc = KAPPA * vo[b * NO + o];
        const float* wr = w_out + o * NR;
        for (int r = 0; r < NR; ++r) acc += zr[r] * wr[r];
        von[o] = acc;
    }
    float mx = von[0];
    #pragma unroll
    for (int o = 1; o < NO; ++o) mx = fmaxf(mx, von[o]);
    float e[NO], s = 0.0f;
    #pragma unroll
    for (int o = 0; o < NO; ++o) { e[o] = expf(von[o] - mx); s += e[o]; }
    float inv = 1.0f / s;
    #pragma unroll
    for (int o = 0; o < NO; ++o) {
        float y = e[o] * inv;
        yo_out[b * NO + o]  = y;
        err_out[b * NO + o] = y - yt[b * NO + o];
    }
}

// ---------------------------------------------------------------------------
// L = err @ w_out [B,R]; Lh = L*h. One thread per (b,r).
// ---------------------------------------------------------------------------
__global__ void learn_sig_kernel(const float* __restrict__ err,
                                 const float* __restrict__ w_out,
                                 const float* __restrict__ h,
                                 float* __restrict__ L_out,
                                 float* __restrict__ Lh_out)
{
    int idx = blockIdx.x * blockDim.x + threadIdx.x;
    if (idx >= NB * NR) return;
    int b = idx / NR, r = idx % NR;
    float acc = 0.0f;
    #pragma unroll
    for (int o = 0; o < NO; ++o) acc += err[b * NO + o] * w_out[o * NR + r];
    L_out[idx]  = acc;
    Lh_out[idx] = acc * h[idx];
}

// ---------------------------------------------------------------------------
// gw_out[o,r] = sum_b err[b,o] * (KAPPA*Fkz[b,r] + z_new[b,r])
// grid = NO blocks, block = NR threads.
// ---------------------------------------------------------------------------
__global__ void gw_out_kernel(const float* __restrict__ err,
                              const float* __restrict__ Fkz,
                              const float* __restrict__ zf,
                              float* __restrict__ gwo)
{
    int o = blockIdx.x, r = threadIdx.x;
    float acc = 0.0f;
    #pragma unroll 4
    for (int b = 0; b < NB; ++b) {
        size_t off = (size_t)b * NR + r;
        acc += err[b * NO + o] * (KAPPA * Fkz[off] + zf[off]);
    }
    gwo[o * NR + r] = acc;
}

// ---------------------------------------------------------------------------
// Streaming HBM-bound reduction: partial[s][i,j] = KAPPA * sum_{b in chunk s}
//   L[b,i] * Fke[b,i,j].  Non-temporal loads (Fke read exactly once).
// ---------------------------------------------------------------------------
template <int NC>
__global__ void stream_reduce_kernel(const float* __restrict__ L,
                                     const float* __restrict__ Fke,
                                     float* __restrict__ partial)
{
    int idx = blockIdx.x * blockDim.x + threadIdx.x;   // i*NC + j
    int s   = blockIdx.y;
    int i = idx / NC, j = idx % NC;
    const int CHUNK = NB / SPLITS;
    int b0 = s * CHUNK;
    const float* p  = Fke + ((size_t)b0 * NR + i) * NC + j;
    const float* lp = L + (size_t)b0 * NR + i;
    float acc = 0.0f;
    #pragma unroll 4
    for (int t = 0; t < CHUNK; ++t) {
        acc += (*lp) * __builtin_nontemporal_load(p);
        p  += (size_t)NR * NC;
        lp += NR;
    }
    partial[(size_t)s * (NR * NC) + idx] = KAPPA * acc;
}

// ---------------------------------------------------------------------------
// gw[i,j] = sum_s partial[s][i,j] + (Lh.T @ Brow)[i,j]   (fp32 WMMA, K = NB)
// A[m=i,k=b] = Lh[b,i] ; B[k=b,n=j] = Brow[b,j]. One wave per 16x16 tile.
// Per-lane loads are lane-coalesced 64B segments (consecutive lanes read
// consecutive i / j columns).
// ---------------------------------------------------------------------------
template <int NC>
__global__ __launch_bounds__(128) void gw_wmma_kernel(
    const float* __restrict__ partial, const float* __restrict__ Lh,
    const float* __restrict__ Brow, float* __restrict__ gw)
{
    const int NTN = NC / 16;
    int wave = blockIdx.x * (blockDim.x >> 5) + (threadIdx.x >> 5);
    int tm = wave / NTN, tn = wave % NTN;
    int lane = threadIdx.x & 31;
    int half = lane >> 4;
    int mr   = lane & 15;
    int n    = tn * 16 + mr;
    int aCol = tm * 16 + mr;

    v8f c;
    #pragma unroll
    for (int g = 0; g < 8; ++g) {
        int m = tm * 16 + g + 8 * half;
        float acc = 0.0f;
        #pragma unroll
        for (int s = 0; s < SPLITS; ++s)
            acc += partial[(size_t)s * (NR * NC) + (size_t)m * NC + n];
        c[g] = acc;
    }
    const float* ap = Lh   + (size_t)(2 * half) * NR + aCol;
    const float* bp = Brow + (size_t)(2 * half) * NC + n;
    #pragma unroll 4
    for (int k0 = 0; k0 < NB; k0 += 4) {
        v2f a, b;
        a.x = ap[0];
        a.y = ap[NR];
        b.x = bp[0];
        b.y = bp[NC];
        ap += 4 * NR;
        bp += 4 * NC;
        c = __builtin_amdgcn_wmma_f32_16x16x4_f32(false, a, false, b,
                                                  (short)0, c, false, false);
    }
    #pragma unroll
    for (int g = 0; g < 8; ++g) {
        int m = tm * 16 + g + 8 * half;
        gw[(size_t)m * NC + n] = c[g];
    }
}

// ---------------------------------------------------------------------------
extern "C" void kernel_launch(void* const* d_in, const int* in_sizes, int n_in,
                              void* d_out, int out_size, void* d_ws, size_t ws_size,
                              hipStream_t stream) {
    const float* x       = (const float*)d_in[0];
    const float* yt      = (const float*)d_in[2];
    const float* w_in    = (const float*)d_in[3];
    const float* w_rec   = (const float*)d_in[4];
    const float* w_out   = (const float*)d_in[5];
    const float* v       = (const float*)d_in[6];
    const float* vo      = (const float*)d_in[7];
    const float* z       = (const float*)d_in[8];
    const float* Faz     = (const float*)d_in[9];
    const float* Fkz     = (const float*)d_in[10];
    const float* Fax     = (const float*)d_in[11];
    const float* Fke_rec = (const float*)d_in[12];
    const float* Fke_in  = (const float*)d_in[13];

    float* out    = (float*)d_out;                 // yo, gw_in, gw_rec, gw_out
    float* yo     = out;                           // 1000*10
    float* gw_in  = out + NB * NO;                 // 256*784
    float* gw_rec = gw_in + NR * NI;               // 256*256
    float* gw_o   = gw_rec + NR * NR;              // 10*256

    float* ws       = (float*)d_ws;
    float* Faz_new  = ws;  ws += NB * NR;
    float* Fax_new  = ws;  ws += NB * NI;
    float* hbuf     = ws;  ws += NB * NR;
    float* zf       = ws;  ws += NB * NR;
    float* errb     = ws;  ws += NB * NO;
    float* Lbuf     = ws;  ws += NB * NR;
    float* Lh       = ws;  ws += NB * NR;
    float* part_rec = ws;  ws += SPLITS * NR * NR;
    float* part_in  = ws;  // SPLITS * NR * NI

    // trace updates (elementwise)
    axpb_kernel<<<(NB * NR + 255) / 256, 256, 0, stream>>>(Faz, z, ALPHA, Faz_new, NB * NR);
    axpb_kernel<<<(NB * NI + 255) / 256, 256, 0, stream>>>(Fax, x, ALPHA, Fax_new, NB * NI);

    // forward membrane update via fp32 WMMA: 63 M-tiles x 16 N-tiles = 1008 waves
    fwd_wmma_kernel<<<1008 / 4, 128, 0, stream>>>(x, z, v, w_in, w_rec, hbuf, zf);

    // readout + softmax + error
    softmax_err_kernel<<<(NB + 255) / 256, 256, 0, stream>>>(zf, vo, w_out, yt, yo, errb);

    // learning signal
    learn_sig_kernel<<<(NB * NR + 255) / 256, 256, 0, stream>>>(errb, w_out, hbuf, Lbuf, Lh);

    // gw_out
    gw_out_kernel<<<NO, NR, 0, stream>>>(errb, Fkz, zf, gw_o);

    // gw_rec: HBM-bound stream (262 MB) then WMMA rank-B correction
    stream_reduce_kernel<NR><<<dim3((NR * NR) / 256, SPLITS), 256, 0, stream>>>(Lbuf, Fke_rec, part_rec);
    gw_wmma_kernel<NR><<<(16 * (NR / 16)) / 4, 128, 0, stream>>>(part_rec, Lh, Faz_new, gw_rec);

    // gw_in: HBM-bound stream (803 MB) then WMMA rank-B correction
    stream_reduce_kernel<NI><<<dim3((NR * NI) / 256, SPLITS), 256, 0, stream>>>(Lbuf, Fke_in, part_in);
    gw_wmma_kernel<NI><<<(16 * (NI / 16)) / 4, 128, 0, stream>>>(part_in, Lh, Fax_new, gw_in);
}